// RPNPostProcessor_78563541778969
// MI455X (gfx1250) — compile-verified
//
#include <hip/hip_runtime.h>
#include <hip/hip_bf16.h>
#include <float.h>

// ---------------- problem constants (match reference) ----------------
constexpr int   N_IMG   = 8;
constexpr int   A_NUM   = 3;
constexpr int   H_DIM   = 336;
constexpr int   W_DIM   = 336;
constexpr int   HW      = H_DIM * W_DIM;        // 112896
constexpr int   M_ELEM  = A_NUM * HW;           // 338688
constexpr int   K_TOP   = 2000;                 // PRE_TOP_K
constexpr int   POST_N  = 1000;                 // POST_NMS_TOP_N
constexpr float NMS_THR   = 0.7f;
constexpr float NEG_VAL   = -1e9f;
constexpr float TO_REMOVE = 1.0f;
constexpr float XFORM_CLIP = 4.135166556742356f;   // log(1000/16)
constexpr float IMG_W = 1333.0f;
constexpr float IMG_H = 800.0f;

constexpr int NK = N_IMG * K_TOP;               // stride between SoA planes

typedef unsigned int u32x4 __attribute__((ext_vector_type(4)));
typedef int          i32x8 __attribute__((ext_vector_type(8)));
typedef int          i32x4 __attribute__((ext_vector_type(4)));

// monotonic key: unsigned order == float order
__device__ __forceinline__ unsigned monokey(float f) {
    unsigned u = __float_as_uint(f);
    return (u & 0x80000000u) ? ~u : (u | 0x80000000u);
}

__device__ __forceinline__ float clampf(float v, float lo, float hi) {
    return fminf(fmaxf(v, lo), hi);
}

// =====================================================================
// Kernel 1: per-image exact top-K select on logits (radix-256, 4 passes,
// LDS histogram) fused with gather + box decode + clip. One WG per image.
// Workspace layout (SoA planes, each NK floats, TDM-friendly):
//   plane 0..5 = x1 | y1 | x2 | y2 | area | score
// =====================================================================
__global__ __launch_bounds__(1024)
void rpn_select_decode_kernel(const float* __restrict__ logits,
                              const float* __restrict__ anchors,
                              const float* __restrict__ breg,
                              float* __restrict__ wsf)
{
    __shared__ unsigned hist[256];
    __shared__ unsigned sPrefix;
    __shared__ unsigned sWant;
    __shared__ unsigned sCntG;
    __shared__ unsigned sCntT;

    const int n   = blockIdx.x;
    const int tid = threadIdx.x;
    const int nthr = blockDim.x;
    const float* lg = logits + (size_t)n * M_ELEM;   // memory order: (a, p)

    unsigned pref = 0;
    unsigned want = (unsigned)K_TOP;

    // ---- 4-pass radix-256 select (descending) ----
    for (int pos = 3; pos >= 0; --pos) {
        for (int b = tid; b < 256; b += nthr) hist[b] = 0u;
        __syncthreads();

        for (int idx = tid; idx < M_ELEM; idx += nthr) {
            // speculative prefetch of the next strip (gfx1250 global_prefetch_b8)
            if (idx + nthr < M_ELEM) __builtin_prefetch(lg + idx + nthr, 0, 1);
            unsigned kk = monokey(lg[idx]);
            bool match = (pos == 3) || ((kk >> (8 * (pos + 1))) == pref);
            if (match) atomicAdd(&hist[(kk >> (8 * pos)) & 255u], 1u);
        }
        __syncthreads();

        if (tid == 0) {
            unsigned acc = 0;
            int b = 255;
            for (; b > 0; --b) {
                unsigned c = hist[b];
                if (acc + c >= want) break;
                acc += c;
            }
            sPrefix = (pref << 8) | (unsigned)b;
            sWant   = want - acc;
        }
        __syncthreads();
        pref = sPrefix;
        want = sWant;
        __syncthreads();   // hist consumed; safe to re-zero next pass
    }

    const unsigned T = pref;                 // key of the K-th largest
    const unsigned R = want;                 // #ties (key == T) to keep, >= 1
    const unsigned G = (unsigned)K_TOP - R;  // #elements with key > T

    if (tid == 0) { sCntG = 0u; sCntT = 0u; }
    __syncthreads();

    float* wx1 = wsf + 0 * NK + n * K_TOP;
    float* wy1 = wsf + 1 * NK + n * K_TOP;
    float* wx2 = wsf + 2 * NK + n * K_TOP;
    float* wy2 = wsf + 3 * NK + n * K_TOP;
    float* war = wsf + 4 * NK + n * K_TOP;
    float* wsc = wsf + 5 * NK + n * K_TOP;

    // ---- compaction + fused gather/decode/clip ----
    for (int idx = tid; idx < M_ELEM; idx += nthr) {
        float lgv = lg[idx];
        unsigned kk = monokey(lgv);
        int slot = -1;
        if (kk > T) {
            slot = (int)atomicAdd(&sCntG, 1u);
        } else if (kk == T) {
            unsigned t = atomicAdd(&sCntT, 1u);
            if (t < R) slot = (int)(G + t);
        }
        if (slot >= 0) {
            // memory index -> score/anchor index m (H,W,A ordering)
            int a = idx / HW;
            int p = idx - a * HW;
            int m = p * A_NUM + a;

            float score = 1.0f / (1.0f + __expf(-lgv));

            float4 anc = ((const float4*)anchors)[(size_t)n * M_ELEM + m];
            const float* rb = breg + ((size_t)(n * A_NUM + a) * 4) * HW + p;
            float dx = rb[0];
            float dy = rb[HW];
            float dw = fminf(rb[2 * HW], XFORM_CLIP);
            float dh = fminf(rb[3 * HW], XFORM_CLIP);

            float w0 = anc.z - anc.x + TO_REMOVE;
            float h0 = anc.w - anc.y + TO_REMOVE;
            float cx = anc.x + 0.5f * w0;
            float cy = anc.y + 0.5f * h0;
            float pcx = dx * w0 + cx;
            float pcy = dy * h0 + cy;
            float pw  = __expf(dw) * w0;
            float ph  = __expf(dh) * h0;

            float x1 = clampf(pcx - 0.5f * pw,        0.0f, IMG_W - 1.0f);
            float y1 = clampf(pcy - 0.5f * ph,        0.0f, IMG_H - 1.0f);
            float x2 = clampf(pcx + 0.5f * pw - 1.0f, 0.0f, IMG_W - 1.0f);
            float y2 = clampf(pcy + 0.5f * ph - 1.0f, 0.0f, IMG_H - 1.0f);

            bool keep = ((x2 - x1 + TO_REMOVE) >= 0.0f) &&
                        ((y2 - y1 + TO_REMOVE) >= 0.0f);

            wx1[slot] = x1;
            wy1[slot] = y1;
            wx2[slot] = x2;
            wy2[slot] = y2;
            war[slot] = (x2 - x1 + TO_REMOVE) * (y2 - y1 + TO_REMOVE);
            wsc[slot] = keep ? score : NEG_VAL;
        }
    }
}

// =====================================================================
// Kernel 2: serial NMS, one WG (1024 threads = 32 waves) per image.
// The entire 6x2000-float SoA candidate block is DMA'd into LDS with a
// single TENSOR_LOAD_TO_LDS (TDM) instruction: a 2D tile, row length
// K_TOP elements, 6 rows, row stride NK elements, 4-byte data.
// Argmax via wave32 shuffle reduction; IoU sweep over LDS.
// =====================================================================
__global__ __launch_bounds__(1024)
void rpn_nms_kernel(const float* __restrict__ wsf,
                    float* __restrict__ out)
{
    __shared__ __align__(16) float smem[6 * K_TOP];   // x1|y1|x2|y2|area|live
    __shared__ float wV[32];
    __shared__ int   wI[32];
    __shared__ int   sJ;
    __shared__ float sSj, sBx1, sBy1, sBx2, sBy2, sAj;

    float* sx1 = smem + 0 * K_TOP;
    float* sy1 = smem + 1 * K_TOP;
    float* sx2 = smem + 2 * K_TOP;
    float* sy2 = smem + 3 * K_TOP;
    float* sar = smem + 4 * K_TOP;
    float* slv = smem + 5 * K_TOP;

    const int n    = blockIdx.x;
    const int tid  = threadIdx.x;
    const int nthr = blockDim.x;           // 1024
    const int lane = tid & 31;             // wave32
    const int wid  = tid >> 5;

    // ---- TDM: one DMA for the whole 48KB candidate block ----
    if (tid == 0) {
        unsigned long long ga =
            (unsigned long long)(const void*)(wsf + (size_t)n * K_TOP);
        unsigned lds_off = (unsigned)(unsigned long long)(void*)smem;

        // D# group 0: count=1 | lds_addr | global_addr | type=2
        u32x4 g0;
        g0.x = 1u;                                   // count=1, user desc
        g0.y = lds_off;                              // lds_addr [63:32]
        g0.z = (unsigned)(ga & 0xffffffffull);       // global_addr lo
        g0.w = (unsigned)((ga >> 32) & 0x1ffffffull) // global_addr hi [120:96]
             | (2u << 30);                           // type=2 ("image")

        // D# group 1: data_size=4B, tensor 2000x6, tile 2000x6, stride NK
        i32x8 g1;
        g1[0] = (int)(2u << 16);                     // data_size=2 (4 bytes)
        g1[1] = (int)(((unsigned)K_TOP & 0xffffu) << 16); // tensor_dim0 lo16
        g1[2] = (int)(6u << 16);                     // tensor_dim0 hi=0 | tensor_dim1=6
        g1[3] = (int)((unsigned)K_TOP << 16);        // tensor_dim1 hi=0 | tile_dim0
        g1[4] = 6;                                   // tile_dim1=6, tile_dim2=0
        g1[5] = NK;                                  // tensor_dim0_stride lo32
        g1[6] = 0;                                   // stride hi | dim1_stride lo
        g1[7] = 0;

        i32x4 g2 = {0, 0, 0, 0};
        i32x4 g3 = {0, 0, 0, 0};
        i32x8 g4 = {0, 0, 0, 0, 0, 0, 0, 0};

        // clang-23 / therock-10.0 lane: 6-arg form
        __builtin_amdgcn_tensor_load_to_lds(g0, g1, g2, g3, g4, 0);
        __builtin_amdgcn_s_wait_tensorcnt(0);
    }
    __syncthreads();

    for (int t = 0; t < POST_N; ++t) {
        // ---- argmax over live scores ----
        float bv = -FLT_MAX;
        int   bi = 0x7fffffff;
        for (int i = tid; i < K_TOP; i += nthr) {
            float v = slv[i];
            if (v > bv || (v == bv && i < bi)) { bv = v; bi = i; }
        }
        #pragma unroll
        for (int off = 16; off > 0; off >>= 1) {
            float ov = __shfl_xor(bv, off, 32);
            int   oi = __shfl_xor(bi, off, 32);
            if (ov > bv || (ov == bv && oi < bi)) { bv = ov; bi = oi; }
        }
        if (lane == 0) { wV[wid] = bv; wI[wid] = bi; }
        __syncthreads();
        if (wid == 0) {
            bv = wV[lane];
            bi = wI[lane];
            #pragma unroll
            for (int off = 16; off > 0; off >>= 1) {
                float ov = __shfl_xor(bv, off, 32);
                int   oi = __shfl_xor(bi, off, 32);
                if (ov > bv || (ov == bv && oi < bi)) { bv = ov; bi = oi; }
            }
            if (lane == 0) {
                sJ = bi; sSj = bv;
                sBx1 = sx1[bi]; sBy1 = sy1[bi];
                sBx2 = sx2[bi]; sBy2 = sy2[bi];
                sAj  = sar[bi];
            }
        }
        __syncthreads();

        const int   j   = sJ;
        const float sj  = sSj;
        const float jx1 = sBx1, jy1 = sBy1, jx2 = sBx2, jy2 = sBy2, aj = sAj;

        // ---- suppress via IoU ----
        for (int i = tid; i < K_TOP; i += nthr) {
            float xx1 = fmaxf(jx1, sx1[i]);
            float yy1 = fmaxf(jy1, sy1[i]);
            float xx2 = fminf(jx2, sx2[i]);
            float yy2 = fminf(jy2, sy2[i]);
            float iw  = fmaxf(xx2 - xx1 + TO_REMOVE, 0.0f);
            float ih  = fmaxf(yy2 - yy1 + TO_REMOVE, 0.0f);
            float inter = iw * ih;
            float iou = inter / (aj + sar[i] - inter);
            if (iou > NMS_THR) slv[i] = NEG_VAL;
        }
        if (tid == 0) {
            slv[j] = NEG_VAL;
            bool valid = sj > NEG_VAL * 0.5f;
            float* o = out + ((size_t)n * POST_N + t) * 5;
            o[0] = valid ? jx1 : 0.0f;
            o[1] = valid ? jy1 : 0.0f;
            o[2] = valid ? jx2 : 0.0f;
            o[3] = valid ? jy2 : 0.0f;
            o[4] = valid ? sj  : 0.0f;
        }
        __syncthreads();
    }
}

// =====================================================================
extern "C" void kernel_launch(void* const* d_in, const int* in_sizes, int n_in,
                              void* d_out, int out_size, void* d_ws, size_t ws_size,
                              hipStream_t stream)
{
    const float* anchors = (const float*)d_in[0];  // [N, M, 4]
    const float* logits  = (const float*)d_in[1];  // [N, A, H, W]
    const float* breg    = (const float*)d_in[2];  // [N, A*4, H, W]
    float* out = (float*)d_out;                    // [N, POST_N, 5]

    float* wsf = (float*)d_ws;                     // 6 SoA planes of NK floats

    rpn_select_decode_kernel<<<dim3(N_IMG), dim3(1024), 0, stream>>>(
        logits, anchors, breg, wsf);

    rpn_nms_kernel<<<dim3(N_IMG), dim3(1024), 0, stream>>>(wsf, out);
}